// Fractal1D_50912542327442
// MI455X (gfx1250) — compile-verified
//
#include <hip/hip_runtime.h>
#include <math.h>

typedef __attribute__((ext_vector_type(2))) float v2f;
typedef __attribute__((ext_vector_type(8))) float v8f;

#define MDIM   128
#define STRIDE 132                 // padded row stride (floats) to avoid LDS bank conflicts
#define WAVES_PER_BLOCK 8
#define ROWS_PER_WAVE   16
#define WBUF   (ROWS_PER_WAVE * STRIDE)   // one w tile: 16 x 132 floats

// LDS layout (float offsets)
#define LT_OFF    0
#define RT_OFF    (MDIM * STRIDE)                       // 16896
#define SPLIT_OFF (2 * MDIM * STRIDE)                   // 33792
#define VAL_OFF   (SPLIT_OFF + MDIM)                    // 33920
#define W_OFF     (VAL_OFF + MDIM)                      // 34048
#define T_OFF     (W_OFF + WAVES_PER_BLOCK * 2 * WBUF)  // 67840
#define LDS_FLOATS (T_OFF + WAVES_PER_BLOCK * 16)       // 67968 floats = 271872 B

__global__ void Fractal1D_kernel(const float* __restrict__ x,
                                 const float* __restrict__ spp,
                                 const float* __restrict__ vp,
                                 const float* __restrict__ Lm,
                                 const float* __restrict__ Rm,
                                 const int*   __restrict__ md,
                                 float* __restrict__ out)
{
    extern __shared__ float lds[];
    const int tid   = threadIdx.x;
    const int lane  = tid & 31;
    const int wave  = tid >> 5;
    const int depth = md[0];

    // ---- Stage L^T and R^T into LDS (transposed so B-fragments are contiguous k-pairs)
    for (int idx = tid; idx < MDIM * MDIM; idx += blockDim.x) {
        const int k = idx >> 7;      // row of L (k index)
        const int n = idx & 127;     // col of L (n index)
        lds[LT_OFF + n * STRIDE + k] = Lm[idx];
        lds[RT_OFF + n * STRIDE + k] = Rm[idx];
    }
    // ---- Stage transformed split points and values
    if (tid < MDIM) {
        const float z = 4.0f * spp[tid] - 2.0f;
        lds[SPLIT_OFF + tid] = 1.0f / (1.0f + __expf(-z));   // sigmoid(4p-2)
        lds[VAL_OFF + tid]   = vp[tid & 31] * 3.0f + 1.0f;   // tile(values*3+1, 4)
    }
    // ---- Init this wave's w tile (buffer 0): w[:,0] = 1, else 0
    float* Wbase = &lds[W_OFF + wave * 2 * WBUF];
    for (int idx = lane; idx < WBUF; idx += 32)
        Wbase[idx] = ((idx % STRIDE) == 0) ? 1.0f : 0.0f;
    __syncthreads();

    const int m = lane & 15;         // point row within the wave's tile
    const int h = lane >> 4;         // lane half (both halves duplicate per-point state)
    const int point = blockIdx.x * (WAVES_PER_BLOCK * ROWS_PER_WAVE)
                      + wave * ROWS_PER_WAVE + m;

    const float xv = x[point];
    float lo = 0.0f, hi = 1.0f;
    float* tArr = &lds[T_OFF + wave * 16];

    int cur = 0;
    for (int d = 0; d < depth; ++d) {
        const float* Wc = Wbase + cur * WBUF;
        float*       Wn = Wbase + (cur ^ 1) * WBUF;

        // ---- s = lo + (w . split) * (hi - lo) : each lane-half dots 64 elems
        {
            const float* wrow = Wc + m * STRIDE + h * 64;
            const float* sp   = &lds[SPLIT_OFF + h * 64];
            float p = 0.0f;
            #pragma unroll 8
            for (int k = 0; k < 64; ++k) p += wrow[k] * sp[k];
            const float dot = p + __shfl_xor(p, 16, 32);
            const float s   = lo + dot * (hi - lo);
            const float t   = 1.0f / (1.0f + __expf((s - xv) * 10.0f)); // sigmoid((x-s)/0.1)
            const float nlo = (1.0f - t) * lo + t * s;
            hi = (1.0f - t) * s + t * hi;
            lo = nlo;
            if (h == 0) tArr[m] = t;   // broadcast t per row for the blend phase
        }

        // ---- Hoist this lane's 8 row-t values into registers (LDS is in-order per wave)
        float tv[8], omt[8];
        #pragma unroll
        for (int r = 0; r < 8; ++r) {
            tv[r]  = tArr[r + 8 * h];
            omt[r] = 1.0f - tv[r];
        }

        // ---- w' = (1-t) row* (w @ L) + t row* (w @ R), tile by tile via f32 WMMA
        for (int jb = 0; jb < 8; ++jb) {
            v8f accL = {0.0f, 0.0f, 0.0f, 0.0f, 0.0f, 0.0f, 0.0f, 0.0f};
            v8f accR = {0.0f, 0.0f, 0.0f, 0.0f, 0.0f, 0.0f, 0.0f, 0.0f};
            // fragment element for lane: (row = m, k = 2h + {0,1}) -> one 8B LDS read
            const float* arow = Wc + m * STRIDE + 2 * h;
            const float* lcol = &lds[LT_OFF + (jb * 16 + m) * STRIDE + 2 * h];
            const float* rcol = &lds[RT_OFF + (jb * 16 + m) * STRIDE + 2 * h];
            #pragma unroll 4
            for (int kb = 0; kb < 32; ++kb) {
                const v2f a  = *(const v2f*)(arow + kb * 4);
                const v2f bl = *(const v2f*)(lcol + kb * 4);
                const v2f br = *(const v2f*)(rcol + kb * 4);
                accL = __builtin_amdgcn_wmma_f32_16x16x4_f32(
                           false, a, false, bl, (short)0, accL, false, false);
                accR = __builtin_amdgcn_wmma_f32_16x16x4_f32(
                           false, a, false, br, (short)0, accR, false, false);
            }
            // C/D layout: VGPR r, lanes 0-15 -> row r; lanes 16-31 -> row r+8
            #pragma unroll
            for (int r = 0; r < 8; ++r) {
                Wn[(r + 8 * h) * STRIDE + jb * 16 + m] =
                    omt[r] * accL[r] + tv[r] * accR[r];
            }
        }
        cur ^= 1;
    }

    // ---- out = w . values
    {
        const float* Wc   = Wbase + cur * WBUF;
        const float* wrow = Wc + m * STRIDE + h * 64;
        const float* vv   = &lds[VAL_OFF + h * 64];
        float p = 0.0f;
        #pragma unroll 8
        for (int k = 0; k < 64; ++k) p += wrow[k] * vv[k];
        const float r = p + __shfl_xor(p, 16, 32);
        if (h == 0) out[point] = r;
    }
}

extern "C" void kernel_launch(void* const* d_in, const int* in_sizes, int n_in,
                              void* d_out, int out_size, void* d_ws, size_t ws_size,
                              hipStream_t stream) {
    (void)n_in; (void)d_ws; (void)ws_size; (void)out_size;
    const float* x   = (const float*)d_in[0];
    const float* spp = (const float*)d_in[1];
    const float* vp  = (const float*)d_in[2];
    const float* Lm  = (const float*)d_in[3];
    const float* Rm  = (const float*)d_in[4];
    const int*   md  = (const int*)d_in[5];
    float* out = (float*)d_out;

    const int n = in_sizes[0];                              // 131072
    const int pointsPerBlock = WAVES_PER_BLOCK * ROWS_PER_WAVE; // 128
    const int blocks = (n + pointsPerBlock - 1) / pointsPerBlock;
    const size_t ldsBytes = (size_t)LDS_FLOATS * sizeof(float); // ~266 KB < 320 KB/WGP

    // Opt in to large dynamic LDS (no-op if already allowed).
    (void)hipFuncSetAttribute((const void*)Fractal1D_kernel,
                              hipFuncAttributeMaxDynamicSharedMemorySize,
                              (int)ldsBytes);

    Fractal1D_kernel<<<blocks, WAVES_PER_BLOCK * 32, ldsBytes, stream>>>(
        x, spp, vp, Lm, Rm, md, out);
}